// SimplicialConvolution_506806141100
// MI455X (gfx1250) — compile-verified
//
#include <hip/hip_runtime.h>

typedef __attribute__((ext_vector_type(2))) float v2f;
typedef __attribute__((ext_vector_type(4))) float v4f;
typedef __attribute__((ext_vector_type(8))) float v8f;

#define CIN  256
#define COUT 256

// ---------------------------------------------------------------------------
// Kernel 1: x_proj = x_src @ W   (fp32, exact, via v_wmma_f32_16x16x4_f32)
//
// W is staged once per workgroup into LDS, pre-swizzled into WMMA B-fragment
// order: for k-step ks, col-tile nt, lane L, the pair (W[k][n], W[k+1][n])
// with k = 4*ks + 2*(L/16), n = 16*nt + L%16 sits at dword index
// ((ks*16+nt)*32 + L)*2. Main-loop B fetch = one ds_load_b64 per fragment;
// the 32 lanes of a wave touch 64 consecutive dwords -> one access per bank,
// conflict-free, and the pair lands even-aligned for WMMA directly.
// Buffer = 256 KB of the 320 KB WGP LDS.
// ---------------------------------------------------------------------------
__global__ __launch_bounds__(256, 1)
void simpconv_proj_gemm(const float* __restrict__ A,
                        const float* __restrict__ W,
                        float* __restrict__ out,
                        int nrows) {
    __shared__ float wfrag[CIN * COUT];   // 256 KB, fragment-swizzled W

    // Swizzle-fill: each thread loads 64 coalesced float4 of W and scatters
    // the 4 elements to their fragment slots (one-time cost).
    for (int i = threadIdx.x; i < (CIN * COUT) / 4; i += 256) {
        const int k  = i >> 6;        // W row
        const int nq = i & 63;        // float4 index within row
        v4f w4 = ((const v4f*)W)[i];
        const int ks = k >> 2;
        const int g  = (k >> 1) & 1;
        const int j  = k & 1;
#pragma unroll
        for (int e = 0; e < 4; ++e) {
            const int n    = nq * 4 + e;
            const int nt   = n >> 4;
            const int m    = n & 15;
            const int lane = (g << 4) | m;
            const int dw   = ((((ks << 4) | nt) << 5) | lane) * 2 + j;
            wfrag[dw] = w4[e];
        }
    }
    __syncthreads();

    const int wave      = threadIdx.x >> 5;
    const int lane      = threadIdx.x & 31;
    const int laneGroup = lane >> 4;     // 0 or 1
    const int laneM     = lane & 15;

    const int rowTile = blockIdx.x * 8 + wave;       // 16 A-rows per wave
    if (rowTile * 16 >= nrows) return;               // wave-uniform: EXEC all-1s

    const float* arow     = A + (size_t)(rowTile * 16 + laneM) * CIN;
    const float* fragLane = wfrag + lane * 2;

    v8f acc[16];
#pragma unroll
    for (int nt = 0; nt < 16; ++nt) {
        v8f z = {0.f, 0.f, 0.f, 0.f, 0.f, 0.f, 0.f, 0.f};
        acc[nt] = z;
    }

    for (int ks = 0; ks < CIN / 4; ++ks) {
        // A fragment 16x4: lane holds (M=laneM, K=4ks+2g+{0,1}) -> b64 load
        v2f a = *(const v2f*)(arow + ks * 4 + laneGroup * 2);
        // Warm the A stream ~16 k-steps ahead (wrapped in-row: always in-bounds)
        __builtin_prefetch(arow + ((ks * 4 + 64) & (CIN - 1)), 0, 3);

#pragma unroll
        for (int nt = 0; nt < 16; ++nt) {
            v2f b = *(const v2f*)(fragLane + (ks * 16 + nt) * 64);
            acc[nt] = __builtin_amdgcn_wmma_f32_16x16x4_f32(
                /*neg_a=*/false, a, /*neg_b=*/false, b,
                /*c_mod=*/(short)0, acc[nt],
                /*reuse_a=*/false, /*reuse_b=*/false);
        }
    }

    // C/D 16x16 f32 layout: VGPR v, lane group g -> M = v + 8*g, N = lane%16
    float* orow = out + (size_t)rowTile * 16 * COUT;
#pragma unroll
    for (int nt = 0; nt < 16; ++nt) {
#pragma unroll
        for (int v = 0; v < 8; ++v) {
            const int m = v + 8 * laneGroup;
            orow[(size_t)m * COUT + nt * 16 + laneM] = acc[nt][v];
        }
    }
}

// ---------------------------------------------------------------------------
// Kernel 2: out[r] += v * x_proj[c]  for each nonzero (r, c, v).
// One wave per nonzero; indices/value fetched via scalar loads
// (readfirstlane-forced uniform); lane l covers channels [l*4, l*4+4) and [128+l*4, 132+l*4):
// two coalesced b128 loads + 8 global_atomic_add_f32 (non-returning).
// ---------------------------------------------------------------------------
__global__ __launch_bounds__(256)
void simpconv_scatter_add(const float* __restrict__ xproj,
                          const float* __restrict__ bvals,
                          const int*   __restrict__ brows,
                          const int*   __restrict__ bcols,
                          float*       __restrict__ out,
                          int nnz) {
    const int wid  = (int)((blockIdx.x * (unsigned)blockDim.x + threadIdx.x) >> 5);
    const int lane = threadIdx.x & 31;
    if (wid >= nnz) return;                          // wave-uniform

    const int swid = __builtin_amdgcn_readfirstlane(wid);  // force SGPR -> s_loads
    const int   r = brows[swid];
    const int   c = bcols[swid];
    const float v = bvals[swid];

    const v4f* src = (const v4f*)(xproj + (size_t)c * COUT);
    float*     dst = out + (size_t)r * COUT;

#pragma unroll
    for (int i = 0; i < 2; ++i) {
        const int q = lane + 32 * i;                 // float4 index 0..63
        v4f m = src[q];
        atomicAdd(dst + q * 4 + 0, v * m.x);
        atomicAdd(dst + q * 4 + 1, v * m.y);
        atomicAdd(dst + q * 4 + 2, v * m.z);
        atomicAdd(dst + q * 4 + 3, v * m.w);
    }
}

// ---------------------------------------------------------------------------
extern "C" void kernel_launch(void* const* d_in, const int* in_sizes, int n_in,
                              void* d_out, int out_size, void* d_ws, size_t ws_size,
                              hipStream_t stream) {
    const float* x_src  = (const float*)d_in[0];   // [N, 256]
    const float* W      = (const float*)d_in[1];   // [256, 256]
    const float* b_vals = (const float*)d_in[2];   // [NNZ]
    const int*   b_rows = (const int*)d_in[3];     // [NNZ]
    const int*   b_cols = (const int*)d_in[4];     // [NNZ]

    const int N   = in_sizes[0] / CIN;
    const int NNZ = in_sizes[2];

    float* xproj = (float*)d_ws;                   // N*COUT*4 bytes of scratch

    // segment_sum baseline: zero the output (harness poisons it)
    hipMemsetAsync(d_out, 0, (size_t)out_size * sizeof(float), stream);

    const int rowTiles = (N + 15) / 16;
    dim3 g1((rowTiles + 7) / 8);
    simpconv_proj_gemm<<<g1, 256, 0, stream>>>(x_src, W, xproj, N);

    dim3 g2((NNZ + 7) / 8);
    simpconv_scatter_add<<<g2, 256, 0, stream>>>(xproj, b_vals, b_rows, b_cols,
                                                 (float*)d_out, NNZ);
}